// NonlinearFilter_24721831756601
// MI455X (gfx1250) — compile-verified
//
#include <hip/hip_runtime.h>
#include <hip/hip_bf16.h>

// Problem constants (from reference)
#define S_ 16
#define B_ 128
#define T_ 200
#define L_ 64
#define R_ 16
#define H_ 256

typedef __attribute__((ext_vector_type(2))) float v2f;
typedef __attribute__((ext_vector_type(8))) float v8f;

__device__ __forceinline__ float softplusf(float x) {
    return (x > 20.0f) ? x : log1pf(expf(x));
}

// ---- WMMA f32 16x16x4 fragment helpers (layouts per CDNA5 ISA) ----
// A (16x4, MxK): lanes 0-15 hold M=lane (K0/K1); lanes 16-31 K2/K3.
__device__ __forceinline__ v2f ldA(const float* s, int ld, int lane) {
    int m = lane & 15, k0 = (lane >> 4) << 1;
    v2f a; a.x = s[m * ld + k0]; a.y = s[m * ld + k0 + 1]; return a;
}
// negated A fragment (for D = C - A*B since f32 WMMA NEG is C-only)
__device__ __forceinline__ v2f ldAn(const float* s, int ld, int lane) {
    int m = lane & 15, k0 = (lane >> 4) << 1;
    v2f a; a.x = -s[m * ld + k0]; a.y = -s[m * ld + k0 + 1]; return a;
}
// A-transposed source: A(m,k) = src[k*ld + m]
__device__ __forceinline__ v2f ldAT(const float* s, int ld, int lane) {
    int m = lane & 15, k0 = (lane >> 4) << 1;
    v2f a; a.x = s[k0 * ld + m]; a.y = s[(k0 + 1) * ld + m]; return a;
}
// B (4x16, KxN): B(k,n) = src[k*ld + n]
__device__ __forceinline__ v2f ldB(const float* s, int ld, int lane) {
    int n = lane & 15, k0 = (lane >> 4) << 1;
    v2f b; b.x = s[k0 * ld + n]; b.y = s[(k0 + 1) * ld + n]; return b;
}
// B-transposed source: B(k,n) = src[n*ld + k]
__device__ __forceinline__ v2f ldBT(const float* s, int ld, int lane) {
    int n = lane & 15, k0 = (lane >> 4) << 1;
    v2f b; b.x = s[n * ld + k0]; b.y = s[n * ld + k0 + 1]; return b;
}
// C/D (16x16): VGPR r holds M = r + 8*(lane/16), N = lane%16
__device__ __forceinline__ v8f ldC(const float* s, int ld, int lane) {
    int n = lane & 15, mb = (lane >> 4) << 3;
    v8f c;
#pragma unroll
    for (int r = 0; r < 8; ++r) c[r] = s[(mb + r) * ld + n];
    return c;
}
__device__ __forceinline__ v8f wmma4(v2f a, v2f b, v8f c) {
    return __builtin_amdgcn_wmma_f32_16x16x4_f32(
        false, a, false, b, (short)0, c, false, false);
}

// =============================================================================
// Fused blocked Cholesky + triangular inverse of a 64x64 SPD matrix in LDS.
//   A (in): SPD matrix; (out): L in lower triangle (upper = garbage)
//   X (out): full L^-1 (zeros above diagonal)
//   WS: >= 8*256 floats of LDS scratch (per-wave 16x16 tiles)
// All O(n^3) work (panels, Schur updates, off-diag inverse) runs on WMMA.
// Force-inlined so address-space inference keeps every access on the LDS path.
// =============================================================================
__device__ __forceinline__ void chol_trinv64(float* A, float* X, float* WS,
                                             int t0, int lane, int wv) {
    // zero strictly-upper *blocks* of X (diag-block upper halves zeroed by inv16)
    for (int e = t0; e < L_ * L_; e += 256) {
        int i = e >> 6, j = e & 63;
        if ((j >> 4) > (i >> 4)) X[e] = 0.0f;
    }

    for (int kb = 0; kb < 4; ++kb) {
        // --- diagonal 16x16 Cholesky + inverse, fully in wave 0 (LDS in-order)
        if (wv == 0) {
            float* Ab = A + (kb * 16) * 64 + kb * 16;
            for (int kc = 0; kc < 16; ++kc) {
                __builtin_amdgcn_wave_barrier();
                float d = sqrtf(Ab[kc * 64 + kc]);
                __builtin_amdgcn_wave_barrier();
                if (lane == 0) Ab[kc * 64 + kc] = d;
                if (lane > kc && lane < 16) Ab[lane * 64 + kc] *= (1.0f / d);
                __builtin_amdgcn_wave_barrier();
                for (int e = lane; e < 256; e += 32) {
                    int i = e >> 4, j = e & 15;
                    if (i > kc && j > kc && j <= i)
                        Ab[i * 64 + j] -= Ab[i * 64 + kc] * Ab[j * 64 + kc];
                }
            }
            __builtin_amdgcn_wave_barrier();
            // X_kk = L_kk^-1 (column-parallel over 16 lanes)
            float* Xb = X + (kb * 16) * 64 + kb * 16;
            if (lane < 16) {
                int c = lane;
                for (int i2 = 0; i2 < 16; ++i2) {
                    if (i2 < c) { Xb[i2 * 64 + c] = 0.0f; continue; }
                    float s2 = (i2 == c) ? 1.0f : 0.0f;
                    for (int j2 = c; j2 < i2; ++j2)
                        s2 -= Ab[i2 * 64 + j2] * Xb[j2 * 64 + c];
                    Xb[i2 * 64 + c] = s2 / Ab[i2 * 64 + i2];
                }
            }
        }
        __syncthreads();

        // --- panels: L_ik = A_ik * X_kk^T (WMMA), wave w -> i = kb+1+w
        int np = 3 - kb;
        if (wv < np) {
            int i = kb + 1 + wv;
            const float* Asrc = A + (i * 16) * 64 + kb * 16;
            v8f c = {};
#pragma unroll
            for (int kk = 0; kk < 4; ++kk) {
                v2f a = ldA(Asrc + kk * 4, 64, lane);
                v2f b = ldBT(X + (kb * 16) * 64 + kb * 16 + kk * 4, 64, lane);
                c = wmma4(a, b, c);
            }
            int n = lane & 15, mb = (lane >> 4) << 3;
#pragma unroll
            for (int r = 0; r < 8; ++r)
                A[(i * 16 + mb + r) * 64 + kb * 16 + n] = c[r];
        }
        __syncthreads();

        // --- Schur updates: A_ij -= L_i,kb * L_j,kb^T (WMMA, negated A frags)
        // pair lists packed as nibbles (i<<2)|j -> pure SALU decode
        int nt = (kb == 0) ? 6 : ((kb == 1) ? 3 : ((kb == 2) ? 1 : 0));
        if (wv < nt) {
            unsigned tab = (kb == 0) ? 0xFED95u * 16u + 0xAu   // (2,2)(1,1)(2,1)(3,1)(3,2)(3,3)
                         : (kb == 1) ? 0xFEAu                  // (2,2)(3,2)(3,3)
                                     : 0xFu;                   // (3,3)
            // note: kb==0 table = nibbles {A,5,9,D,E,F} from low to high
            unsigned nib = (tab >> (4 * wv)) & 15u;
            int i = (int)(nib >> 2), j = (int)(nib & 3u);
            v8f c = ldC(A + (i * 16) * 64 + j * 16, 64, lane);
#pragma unroll
            for (int kk = 0; kk < 4; ++kk) {
                v2f a = ldAn(A + (i * 16) * 64 + kb * 16 + kk * 4, 64, lane);
                v2f b = ldBT(A + (j * 16) * 64 + kb * 16 + kk * 4, 64, lane);
                c = wmma4(a, b, c);
            }
            int n = lane & 15, mb = (lane >> 4) << 3;
#pragma unroll
            for (int r = 0; r < 8; ++r)
                A[(i * 16 + mb + r) * 64 + j * 16 + n] = c[r];
        }
        __syncthreads();
    }

    // --- off-diagonal inverse blocks, by anti-diagonal d = i-j:
    //     X_ij = -X_ii * (sum_{k=j}^{i-1} L_ik X_kj)       (all WMMA)
    for (int d = 1; d < 4; ++d) {
        int cnt = 4 - d;
        if (wv < cnt) {
            int j = wv, i = j + d;
            float* W = WS + wv * 256;              // 16x16 wave-private scratch
            v8f c = {};
            for (int k = j; k < i; ++k) {
#pragma unroll
                for (int kk = 0; kk < 4; ++kk) {
                    v2f a = ldA(A + (i * 16) * 64 + k * 16 + kk * 4, 64, lane);
                    v2f b = ldB(X + (k * 16 + kk * 4) * 64 + j * 16, 64, lane);
                    c = wmma4(a, b, c);
                }
            }
            int n = lane & 15, mb = (lane >> 4) << 3;
#pragma unroll
            for (int r = 0; r < 8; ++r) W[(mb + r) * 16 + n] = c[r];
            __builtin_amdgcn_wave_barrier();       // same-wave LDS: in-order
            v8f c2 = {};
#pragma unroll
            for (int kk = 0; kk < 4; ++kk) {
                v2f a = ldA(X + (i * 16) * 64 + i * 16 + kk * 4, 64, lane);
                v2f b = ldB(W + kk * 4 * 16, 16, lane);
                c2 = wmma4(a, b, c2);
            }
#pragma unroll
            for (int r = 0; r < 8; ++r)
                X[(i * 16 + mb + r) * 64 + j * 16 + n] = -c2[r];
        }
        __syncthreads();
    }
}

// ---- J (64x64) += Kt * Kt^T, Kt is 64x16 row-major --------------------------
__device__ __forceinline__ void kkt_acc(float* J, const float* Ktl, int lane, int wv) {
    for (int r2 = 0; r2 < 2; ++r2) {
        int tt = wv * 2 + r2, ti = tt >> 2, tj = tt & 3;
        v8f c = ldC(J + ti * 16 * L_ + tj * 16, L_, lane);
#pragma unroll
        for (int kk = 0; kk < 4; ++kk) {
            v2f a = ldA (Ktl + ti * 16 * R_ + kk * 4, R_, lane);
            v2f bb = ldBT(Ktl + tj * 16 * R_ + kk * 4, R_, lane);
            c = wmma4(a, bb, c);
        }
        int n = lane & 15, mb = (lane >> 4) << 3;
#pragma unroll
        for (int r = 0; r < 8; ++r)
            J[(ti * 16 + mb + r) * L_ + tj * 16 + n] = c[r];
    }
}

// ---- J = X^T X for 64x64 X ---------------------------------------------------
__device__ __forceinline__ void ata_stage(const float* X, float* J, int lane, int wv) {
    for (int r2 = 0; r2 < 2; ++r2) {
        int tt = wv * 2 + r2, ti = tt >> 2, tj = tt & 3;
        v8f c = {};
#pragma unroll 4
        for (int kk = 0; kk < 16; ++kk) {
            v2f a = ldAT(X + kk * 4 * L_ + ti * 16, L_, lane);
            v2f bb = ldB (X + kk * 4 * L_ + tj * 16, L_, lane);
            c = wmma4(a, bb, c);
        }
        int n = lane & 15, mb = (lane >> 4) << 3;
#pragma unroll
        for (int r = 0; r < 8; ++r)
            J[(ti * 16 + mb + r) * L_ + tj * 16 + n] = c[r];
    }
}

// ---- z = m_f + Wt * Li  (16x64); stores next Zs and z output ----------------
__device__ __forceinline__ void zf_stage(const float* Wtl, const float* Lil,
                                         const float* mfl, float* Zsl,
                                         float* zout, int bb, int tt_, int lane, int wv) {
    if (wv < 4) {
        int tj = wv;
        v8f c = {};
#pragma unroll 4
        for (int kk = 0; kk < 16; ++kk) {
            v2f a = ldA(Wtl + kk * 4, L_, lane);
            v2f bb_ = ldB(Lil + kk * 4 * L_ + tj * 16, L_, lane);
            c = wmma4(a, bb_, c);
        }
        int n = lane & 15, mb = (lane >> 4) << 3;
#pragma unroll
        for (int r = 0; r < 8; ++r) {
            int s = mb + r, col = tj * 16 + n;
            float v = c[r] + mfl[col];
            Zsl[s * L_ + col] = v;
            zout[((size_t)s * B_ + bb) * T_ * L_ + (size_t)tt_ * L_ + col] = v;
        }
    }
}

__global__ __launch_bounds__(256) void nlfilter_kernel(
    const float* __restrict__ kv,  const float* __restrict__ Kmat,
    const float* __restrict__ lQ,  const float* __restrict__ m0,
    const float* __restrict__ lv0, const float* __restrict__ W1,
    const float* __restrict__ b1,  const float* __restrict__ W2,
    const float* __restrict__ b2,  const float* __restrict__ w,
    float* __restrict__ z_out, float* __restrict__ mf_out,
    float* __restrict__ mp_out, float* __restrict__ pf_out,
    float* __restrict__ pp_out)
{
    __shared__ float Zs[S_ * L_];    // z_prev samples (16x64)
    __shared__ float A1s[S_ * H_];   // hidden activations; also chol scratch
    __shared__ float Ms[S_ * L_];    // mean_fn output (16x64)
    __shared__ float Pm[L_ * L_];    // P_p -> Lp (in place)
    __shared__ float Li[L_ * L_];    // Lp^-1, later Lf^-1
    __shared__ float Jm[L_ * L_];    // J_p -> J_f -> Lf (in place)
    __shared__ float Kt[L_ * R_];    // K_t (64x16)
    __shared__ float Wt[S_ * L_];    // w_t for this b (16x64)
    __shared__ float mp[L_], hf[L_], mf[L_], tv[L_], qd[L_], j0v[L_];

    const int b = blockIdx.x;
    const int t0 = threadIdx.x;
    const int lane = t0 & 31, wv = t0 >> 5;

    // ---------------- init (t = 0) ----------------
    if (t0 < L_) {
        qd[t0] = softplusf(lQ[t0]);
        float p0 = softplusf(lv0[t0]);
        j0v[t0] = 1.0f / p0;
        tv[t0] = sqrtf(p0);                       // Pp_chol0 = diag(sqrt(P0))
    }
    for (int e = t0; e < L_ * R_; e += 256)
        Kt[e] = Kmat[(size_t)b * T_ * L_ * R_ + e];          // K[b,0,:,:]
    __syncthreads();

    {   // Jf0 = diag(J0) + K0 K0^T
        for (int e = t0; e < L_ * L_; e += 256) {
            int i = e >> 6, j = e & 63;
            Jm[e] = (i == j) ? j0v[i] : 0.0f;
        }
        size_t base0 = (size_t)b * T_ * L_ * L_;
        for (int e = t0; e < L_ * L_; e += 256) {             // Pp_chol t=0
            int i = e >> 6, j = e & 63;
            pp_out[base0 + e] = (i == j) ? tv[i] : 0.0f;
        }
        __syncthreads();
        kkt_acc(Jm, Kt, lane, wv);
        __syncthreads();
        chol_trinv64(Jm, Li, A1s, t0, lane, wv);              // Lf0, Lf0^-1

        if (t0 < L_) hf[t0] = j0v[t0] * m0[t0] + kv[(size_t)b * T_ * L_ + t0];
        __syncthreads();
        if (t0 < L_) { float s = 0; for (int j = 0; j < 64; ++j) s += Li[t0 * 64 + j] * hf[j]; tv[t0] = s; }
        __syncthreads();
        if (t0 < L_) {
            float s = 0; for (int j = 0; j < 64; ++j) s += Li[j * 64 + t0] * tv[j];
            mf[t0] = s;
            mf_out[((size_t)b * T_) * L_ + t0] = s;
            mp_out[((size_t)b * T_) * L_ + t0] = m0[t0];      // m_p0 = m0
        }
        __syncthreads();
        // Pf_chol0 = Li^T : lanes vary i -> LDS bank-conflict-free reads
        for (int e = t0; e < L_ * L_; e += 256) {
            int j = e >> 6, i = e & 63;
            pf_out[base0 + (size_t)i * 64 + j] = Li[j * 64 + i];
        }
        for (int e = t0; e < S_ * L_; e += 256) {             // w[0,:,b,:]
            int s = e >> 6, l = e & 63;
            Wt[e] = w[(((size_t)0 * S_ + s) * B_ + b) * L_ + l];
        }
        __syncthreads();
        zf_stage(Wt, Li, mf, Zs, z_out, b, 0, lane, wv);      // z_f0 -> Zs
        __syncthreads();
    }

    // ---------------- scan t = 1..T-1 ----------------
    for (int t = 1; t < T_; ++t) {
        for (int e = t0; e < L_ * R_; e += 256)
            Kt[e] = Kmat[((size_t)b * T_ + t) * L_ * R_ + e];
        for (int e = t0; e < S_ * L_; e += 256) {
            int s = e >> 6, l = e & 63;
            Wt[e] = w[(((size_t)t * S_ + s) * B_ + b) * L_ + l];
        }
        __syncthreads();

        // --- MLP layer 1: A1 = tanh(Zs * W1 + b1), 16x256 ---
        for (int r2 = 0; r2 < 2; ++r2) {
            int tj = wv * 2 + r2;
            v8f c = {};
#pragma unroll 4
            for (int kk = 0; kk < 16; ++kk) {
                v2f a  = ldA(Zs + kk * 4, L_, lane);
                v2f bb = ldB(W1 + (size_t)(kk * 4) * H_ + tj * 16, H_, lane);
                c = wmma4(a, bb, c);
            }
            int n = lane & 15, mb = (lane >> 4) << 3;
#pragma unroll
            for (int r = 0; r < 8; ++r) {
                int col = tj * 16 + n;
                A1s[(mb + r) * H_ + col] = tanhf(c[r] + b1[col]);
            }
        }
        __syncthreads();

        // --- MLP layer 2: M = A1 * W2 + b2, 16x64 (waves 0-3) ---
        if (wv < 4) {
            int tj = wv;
            v8f c = {};
#pragma unroll 4
            for (int kk = 0; kk < 64; ++kk) {
                v2f a  = ldA(A1s + kk * 4, H_, lane);
                v2f bb = ldB(W2 + (size_t)(kk * 4) * L_ + tj * 16, L_, lane);
                c = wmma4(a, bb, c);
            }
            int n = lane & 15, mb = (lane >> 4) << 3;
#pragma unroll
            for (int r = 0; r < 8; ++r) {
                int col = tj * 16 + n;
                Ms[(mb + r) * L_ + col] = c[r] + b2[col];
            }
        }
        __syncthreads();

        // --- m_p = mean_s M ---
        if (t0 < L_) {
            float s = 0;
            for (int ss = 0; ss < S_; ++ss) s += Ms[ss * L_ + t0];
            s *= (1.0f / S_);
            mp[t0] = s;
            mp_out[((size_t)b * T_ + t) * L_ + t0] = s;
        }
        __syncthreads();

        // --- P_p = diag(Q) + M^T M / S - mp mp^T ---
        for (int r2 = 0; r2 < 2; ++r2) {
            int tt = wv * 2 + r2, ti = tt >> 2, tj = tt & 3;
            v8f c = {};
#pragma unroll
            for (int kk = 0; kk < 4; ++kk) {
                v2f a  = ldAT(Ms + kk * 4 * L_ + ti * 16, L_, lane);
                v2f bb = ldB (Ms + kk * 4 * L_ + tj * 16, L_, lane);
                c = wmma4(a, bb, c);
            }
            int n = lane & 15, mb = (lane >> 4) << 3;
#pragma unroll
            for (int r = 0; r < 8; ++r) {
                int i = ti * 16 + mb + r, j = tj * 16 + n;
                float v = c[r] * (1.0f / S_) - mp[i] * mp[j];
                if (i == j) v += qd[i];
                Pm[i * L_ + j] = v;
            }
        }
        __syncthreads();

        // --- Lp = chol(P_p), Li = Lp^-1 (blocked, WMMA) ---
        chol_trinv64(Pm, Li, A1s, t0, lane, wv);
        {
            size_t base = ((size_t)b * T_ + t) * L_ * L_;
            for (int e = t0; e < L_ * L_; e += 256) {
                int i = e >> 6, j = e & 63;
                pp_out[base + e] = (j <= i) ? Pm[i * 64 + j] : 0.0f;
            }
        }

        // --- h_f = Li^T (Li mp) + k_t ---
        if (t0 < L_) { float s = 0; for (int j = 0; j < 64; ++j) s += Li[t0 * 64 + j] * mp[j]; tv[t0] = s; }
        __syncthreads();
        if (t0 < L_) {
            float s = 0; for (int j = 0; j < 64; ++j) s += Li[j * 64 + t0] * tv[j];
            hf[t0] = s + kv[((size_t)b * T_ + t) * L_ + t0];
        }
        __syncthreads();

        // --- J_f = Li^T Li + K_t K_t^T; Lf = chol(J_f); Li = Lf^-1 ---
        ata_stage(Li, Jm, lane, wv);
        __syncthreads();
        kkt_acc(Jm, Kt, lane, wv);
        __syncthreads();
        chol_trinv64(Jm, Li, A1s, t0, lane, wv);

        // --- m_f = Li^T (Li h_f) ---
        if (t0 < L_) { float s = 0; for (int j = 0; j < 64; ++j) s += Li[t0 * 64 + j] * hf[j]; tv[t0] = s; }
        __syncthreads();
        if (t0 < L_) {
            float s = 0; for (int j = 0; j < 64; ++j) s += Li[j * 64 + t0] * tv[j];
            mf[t0] = s;
            mf_out[((size_t)b * T_ + t) * L_ + t0] = s;
        }
        __syncthreads();

        // --- P_f_chol out = Li^T (bank-conflict-free LDS reads) ---
        {
            size_t base = ((size_t)b * T_ + t) * L_ * L_;
            for (int e = t0; e < L_ * L_; e += 256) {
                int j = e >> 6, i = e & 63;
                pf_out[base + (size_t)i * 64 + j] = Li[j * 64 + i];
            }
        }

        // --- z_f = m_f + Wt * Li  -> next Zs + output ---
        zf_stage(Wt, Li, mf, Zs, z_out, b, t, lane, wv);
        __syncthreads();
    }
}

extern "C" void kernel_launch(void* const* d_in, const int* in_sizes, int n_in,
                              void* d_out, int out_size, void* d_ws, size_t ws_size,
                              hipStream_t stream) {
    (void)in_sizes; (void)n_in; (void)out_size; (void)d_ws; (void)ws_size;
    const float* kv  = (const float*)d_in[0];
    const float* K   = (const float*)d_in[1];
    const float* lQ  = (const float*)d_in[2];
    const float* m0  = (const float*)d_in[3];
    const float* lv0 = (const float*)d_in[4];
    const float* W1  = (const float*)d_in[5];
    const float* b1  = (const float*)d_in[6];
    const float* W2  = (const float*)d_in[7];
    const float* b2  = (const float*)d_in[8];
    const float* w   = (const float*)d_in[9];
    float* out = (float*)d_out;

    size_t o_z  = 0;
    size_t o_mf = (size_t)S_ * B_ * T_ * L_;
    size_t o_mp = o_mf + (size_t)B_ * T_ * L_;
    size_t o_pf = o_mp + (size_t)B_ * T_ * L_;
    size_t o_pp = o_pf + (size_t)B_ * T_ * (size_t)L_ * L_;

    nlfilter_kernel<<<B_, 256, 0, stream>>>(
        kv, K, lQ, m0, lv0, W1, b1, W2, b2, w,
        out + o_z, out + o_mf, out + o_mp, out + o_pf, out + o_pp);
}